// RunwayAlignmentLoss_16681652978423
// MI455X (gfx1250) — compile-verified
//
#include <hip/hip_runtime.h>
#include <hip/hip_bf16.h>
#include <stdint.h>

#define B_SAMPLES 65536
#define H_LEN     120
#define N_FINAL   4
#define EPS       1e-06f
#define SCALE     1.0f

typedef __attribute__((ext_vector_type(2))) float v2f;
typedef __attribute__((ext_vector_type(8))) float v8f;

// ---------------------------------------------------------------------------
// Stage 1: one thread per sample. Count valid entries via popcount on the
// bool mask row (bytes are 0/1, row stride 120 bytes = 15 aligned u64 loads),
// gather the last 4 (x,y) points, compute sum of (1 - cos)^2 / 3, then a
// deterministic LDS tree reduction -> one partial per block.
// ---------------------------------------------------------------------------
__global__ __launch_bounds__(256)
void runway_partial_kernel(const float* __restrict__ pred,
                           const unsigned char* __restrict__ mask,
                           const float* __restrict__ runway,
                           float* __restrict__ partials) {
    const int tid = threadIdx.x;
    const int b   = blockIdx.x * 256 + tid;

    // --- count ones in mask row (valid = H - ones) ---
    const uint64_t* mrow = (const uint64_t*)(mask + (size_t)b * H_LEN);
    int ones = 0;
#pragma unroll
    for (int i = 0; i < H_LEN / 8; ++i) {
        ones += __popcll(mrow[i]);
    }
    const int valid = H_LEN - ones;

    // --- gather last N_FINAL (x,y) points of pred_abs ---
    const float* p = pred + ((size_t)b * H_LEN + (size_t)(valid - N_FINAL)) * 3;
    float px[N_FINAL], py[N_FINAL];
#pragma unroll
    for (int j = 0; j < N_FINAL; ++j) {
        px[j] = p[j * 3 + 0];
        py[j] = p[j * 3 + 1];
    }

    // --- normalized runway direction (components 2,3) ---
    float rx = runway[(size_t)b * 4 + 2];
    float ry = runway[(size_t)b * 4 + 3];
    {
        const float rn = sqrtf(rx * rx + ry * ry) + EPS;
        const float inv = 1.0f / rn;
        rx *= inv;
        ry *= inv;
    }

    // --- 3 segment directions, cosine vs runway, (1-cos)^2 ---
    float s = 0.0f;
#pragma unroll
    for (int j = 0; j < N_FINAL - 1; ++j) {
        const float dx = px[j + 1] - px[j];
        const float dy = py[j + 1] - py[j];
        const float dn = sqrtf(dx * dx + dy * dy) + EPS;
        const float cs = (dx * rx + dy * ry) / dn;
        const float t  = 1.0f - cs;
        s += t * t;
    }
    const float per_sample = s * (1.0f / 3.0f);

    // --- deterministic block tree reduction ---
    __shared__ float red[256];
    red[tid] = per_sample;
    __syncthreads();
#pragma unroll
    for (int stride = 128; stride > 0; stride >>= 1) {
        if (tid < stride) red[tid] += red[tid + stride];
        __syncthreads();
    }
    if (tid == 0) partials[blockIdx.x] = red[0];
}

// ---------------------------------------------------------------------------
// Stage 2: single wave32. Reduce the 256 block partials with
// V_WMMA_F32_16X16X4_F32 using a ones B-matrix: D = A(16x4 data) x ones + C.
// After 4 accumulating WMMAs over 64-element chunks, column n==0 of D holds
// the 16 row-sums: lane 0 has m=0..7 (VGPR 0..7), lane 16 has m=8..15.
// A-matrix f32 16x4 layout: element (m,k) lives in lane m + 16*(k>=2),
// VGPR k%2  =>  lane L, VGPR v holds flat index (L&15)*4 + (L>>4)*2 + v.
// ---------------------------------------------------------------------------
__global__ __launch_bounds__(32)
void runway_final_kernel(const float* __restrict__ partials,
                         float* __restrict__ out) {
    const int lane  = threadIdx.x;           // 0..31, EXEC all ones
    const int m     = lane & 15;
    const int kbase = (lane >> 4) * 2;

    v2f ones;
    ones.x = 1.0f;
    ones.y = 1.0f;

    v8f acc = {};
#pragma unroll
    for (int c = 0; c < 4; ++c) {
        const float* chunk = partials + c * 64;
        v2f a;
        a.x = chunk[m * 4 + kbase + 0];
        a.y = chunk[m * 4 + kbase + 1];
        // 8 args: (neg_a, A, neg_b, B, c_mod, C, reuse_a, reuse_b)
        acc = __builtin_amdgcn_wmma_f32_16x16x4_f32(
            /*neg_a=*/false, a, /*neg_b=*/false, ones,
            /*c_mod=*/(short)0, acc, /*reuse_a=*/false, /*reuse_b=*/false);
    }

    // Per-lane partial over this lane's 8 row entries of column n = lane&15.
    float s = acc[0] + acc[1] + acc[2] + acc[3] +
              acc[4] + acc[5] + acc[6] + acc[7];

    // Column 0 lives in lanes 0 (m=0..7) and 16 (m=8..15).
    const float total = __shfl(s, 0, 32) + __shfl(s, 16, 32);

    if (lane == 0) {
        out[0] = total * (SCALE / (float)B_SAMPLES);
    }
}

extern "C" void kernel_launch(void* const* d_in, const int* in_sizes, int n_in,
                              void* d_out, int out_size, void* d_ws, size_t ws_size,
                              hipStream_t stream) {
    const float*         pred   = (const float*)d_in[0];
    // d_in[1] = target_abs: unused by the reference computation.
    const unsigned char* mask   = (const unsigned char*)d_in[2];
    const float*         runway = (const float*)d_in[3];
    float*               out    = (float*)d_out;
    float*               partials = (float*)d_ws;   // 256 floats

    runway_partial_kernel<<<B_SAMPLES / 256, 256, 0, stream>>>(pred, mask, runway, partials);
    runway_final_kernel<<<1, 32, 0, stream>>>(partials, out);
}